// SinkhornEval_38800734552855
// MI455X (gfx1250) — compile-verified
//
#include <hip/hip_runtime.h>
#include <math.h>

// ---------------------------------------------------------------------------
// Sinkhorn divergence (quaternion geodesic cost) for MI455X / gfx1250.
//
// Core op: softmin(eps, C, h)[i] = -eps * logsumexp_j( h[j] - C[i,j]/eps )
// with C[i,j] = 2*acos(min(|<qa_i, qb_j>|, 1-1e-6)).
//
// The 4-wide quaternion dot is computed with V_WMMA_F32_16X16X4_F32
// (16x4 A times 4x16 B -> 16x16 f32 tile) — K=4 matches D=4 exactly.
// Each wave owns a 16-row strip and streams 128 column tiles, keeping a
// flash-attention-style online logsumexp (running max m, running sum s)
// per D-register slot, then merges across the 16 lanes of each half-wave.
// x/y (512 KB total) stay resident in L2; columns + h are staged in LDS.
// ---------------------------------------------------------------------------

typedef float v2f __attribute__((ext_vector_type(2)));
typedef float v8f __attribute__((ext_vector_type(8)));

#define BB 8
#define PP 2048
#define ROWS_PER_WG 128   // 8 waves * 16 rows
#define NTILES (PP / 16)
#define FLT_NEG_BIG (-3.402823466e38f)

// mode: 0 = init (no potential, overwrite), 1 = anneal (pot, 0.5 average),
//       2 = final extrapolation (pot, overwrite)
__global__ __launch_bounds__(256)
void sinkhorn_softmin_pass(
    const float* __restrict__ X, const float* __restrict__ Y,
    const float* __restrict__ aLog, const float* __restrict__ bLog,
    const float* __restrict__ f_ba_o, const float* __restrict__ g_ab_o,
    const float* __restrict__ f_aa_o, const float* __restrict__ g_bb_o,
    float* __restrict__ f_ba_n, float* __restrict__ g_ab_n,
    float* __restrict__ f_aa_n, float* __restrict__ g_bb_n,
    float eps, float inv_eps, int mode)
{
    __shared__ float sh_h[PP];       // h[j] = log-weight + potential/eps
    __shared__ float sh_q[PP * 4];   // column quaternions

    const int tid  = threadIdx.x;
    const int blk  = blockIdx.x;
    const int job  = blk >> 7;       // 0..3
    const int sub  = blk & 127;
    const int b    = sub >> 4;       // 0..7 batch
    const int rt   = sub & 15;       // 0..15 row tile of 128
    const int row0 = rt * ROWS_PER_WG;

    // job table: {rows qa, cols qb, base log-weights, potential, old, out}
    const float* qa; const float* qb; const float* hb; const float* pot;
    const float* oldp; float* outp;
    switch (job) {
        case 0:  qa=X; qb=Y; hb=bLog; pot=g_ab_o; oldp=f_ba_o; outp=f_ba_n; break; // ft_ba
        case 1:  qa=Y; qb=X; hb=aLog; pot=f_ba_o; oldp=g_ab_o; outp=g_ab_n; break; // gt_ab
        case 2:  qa=X; qb=X; hb=aLog; pot=f_aa_o; oldp=f_aa_o; outp=f_aa_n; break; // f_aa
        default: qa=Y; qb=Y; hb=bLog; pot=g_bb_o; oldp=g_bb_o; outp=g_bb_n; break; // g_bb
    }
    const int   use_pot = (mode != 0);
    const float w_old   = (mode == 1) ? 0.5f : 0.0f;
    const float w_new   = (mode == 1) ? 0.5f : 1.0f;

    // ---- stage h and column quaternions into LDS (cooperative) ----
    const float* hbB  = hb  + (size_t)b * PP;
    const float* potB = pot + (size_t)b * PP;
    for (int j = tid; j < PP; j += 256)
        sh_h[j] = use_pot ? fmaf(potB[j], inv_eps, hbB[j]) : hbB[j];
    const float* qbB = qb + (size_t)b * PP * 4;
    for (int k = tid; k < PP * 4; k += 256)
        sh_q[k] = qbB[k];
    __syncthreads();

    const int wave = tid >> 5;
    const int lane = tid & 31;
    const int half = lane >> 4;       // 0: K=0,1  1: K=2,3 (ISA 16x4 A layout)
    const int kh   = half * 2;
    const int Ml   = lane & 15;

    // ---- A operand: 16 row quaternions for this wave (loaded once) ----
    const int rowsBase = row0 + wave * 16;
    const float* qaR = qa + ((size_t)b * PP + rowsBase + Ml) * 4;
    v2f A; A.x = qaR[kh]; A.y = qaR[kh + 1];

    float m[8], s[8];
    #pragma unroll
    for (int r = 0; r < 8; ++r) { m[r] = FLT_NEG_BIG; s[r] = 0.0f; }

    // ---- stream 128 column tiles: WMMA dot + online logsumexp ----
    for (int t = 0; t < NTILES; ++t) {
        const int col = t * 16 + Ml;
        v2f Bm; Bm.x = sh_q[col * 4 + kh]; Bm.y = sh_q[col * 4 + kh + 1];
        v8f c = {};
        v8f d = __builtin_amdgcn_wmma_f32_16x16x4_f32(
                    false, A, false, Bm, (short)0, c, false, false);
        const float hv = sh_h[col];
        #pragma unroll
        for (int r = 0; r < 8; ++r) {
            float u  = fminf(fabsf(d[r]), 0.999999f);   // clip(|dot|, 0, 1-1e-6)
            float Cv = 2.0f * acosf(u);                 // geodesic distance
            float v  = fmaf(-Cv, inv_eps, hv);          // h[j] - C/eps
            float dd = v - m[r];
            float e  = __expf(-fabsf(dd));              // one exp per element
            if (dd > 0.0f) { s[r] = fmaf(s[r], e, 1.0f); m[r] = v; }
            else           { s[r] = s[r] + e; }
        }
    }

    // ---- merge logsumexp state across the 16 lanes of each half-wave ----
    #pragma unroll
    for (int k = 1; k < 16; k <<= 1) {
        #pragma unroll
        for (int r = 0; r < 8; ++r) {
            float mo = __shfl_xor(m[r], k, 32);
            float so = __shfl_xor(s[r], k, 32);
            float Mx = fmaxf(m[r], mo);
            s[r] = fmaf(s[r], __expf(m[r] - Mx), so * __expf(mo - Mx));
            m[r] = Mx;
        }
    }

    // lanes 0-15 now hold rows M=0..7 (slot r), lanes 16-31 rows M=8..15
    if (Ml == 0) {
        const size_t base = (size_t)b * PP + rowsBase + half * 8;
        #pragma unroll
        for (int r = 0; r < 8; ++r) {
            float res = -eps * (m[r] + __logf(s[r]));
            float o   = (w_old != 0.0f) ? fmaf(w_old, oldp[base + r], w_new * res)
                                        : res;
            outp[base + r] = o;
        }
    }
}

__global__ void sinkhorn_init_logs(const float* __restrict__ wx,
                                   const float* __restrict__ wy,
                                   float* __restrict__ aL,
                                   float* __restrict__ bL, int n)
{
    int i = blockIdx.x * blockDim.x + threadIdx.x;
    if (i < n) { aL[i] = __logf(wx[i]); bL[i] = __logf(wy[i]); }
}

__global__ __launch_bounds__(256)
void sinkhorn_reduce_loss(const float* __restrict__ wx, const float* __restrict__ wy,
                          const float* __restrict__ f_ba, const float* __restrict__ f_aa,
                          const float* __restrict__ g_ab, const float* __restrict__ g_bb,
                          float* __restrict__ out, int n)
{
    __shared__ float red[256];
    float acc = 0.0f;
    for (int i = threadIdx.x; i < n; i += 256)
        acc += wx[i] * (f_ba[i] - f_aa[i]) + wy[i] * (g_ab[i] - g_bb[i]);
    red[threadIdx.x] = acc;
    __syncthreads();
    for (int st = 128; st > 0; st >>= 1) {
        if (threadIdx.x < st) red[threadIdx.x] += red[threadIdx.x + st];
        __syncthreads();
    }
    if (threadIdx.x == 0) out[0] = red[0] / (float)BB;
}

extern "C" void kernel_launch(void* const* d_in, const int* in_sizes, int n_in,
                              void* d_out, int out_size, void* d_ws, size_t ws_size,
                              hipStream_t stream)
{
    const float* x  = (const float*)d_in[0];   // (8,2048,4)
    const float* y  = (const float*)d_in[1];   // (8,2048,4)
    const float* wx = (const float*)d_in[2];   // (8,2048)
    const float* wy = (const float*)d_in[3];   // (8,2048)
    float* out = (float*)d_out;                // scalar loss

    const int N = BB * PP;
    float* ws = (float*)d_ws;
    float* aL = ws;
    float* bL = ws + N;
    float* f_ba[2] = { ws + 2*N, ws + 3*N };
    float* g_ab[2] = { ws + 4*N, ws + 5*N };
    float* f_aa[2] = { ws + 6*N, ws + 7*N };
    float* g_bb[2] = { ws + 8*N, ws + 9*N };

    sinkhorn_init_logs<<<(N + 255) / 256, 256, 0, stream>>>(wx, wy, aL, bL, N);

    // geomloss epsilon_schedule(p=2, diameter=3.15, blur=0.01, scaling=0.5)
    double epsv[24]; int ne = 0;
    const double p = 2.0, diam = 3.15, blur = 0.01, scal = 0.5;
    epsv[ne++] = pow(diam, p);
    for (double e = p * log(diam); e > p * log(blur); e += p * log(scal))
        epsv[ne++] = exp(e);
    epsv[ne++] = pow(blur, p);

    int cur = 0;
    // init pass: f_aa/g_bb/g_ab/f_ba = softmin(eps0, ..., log-weights)
    {
        float e = (float)epsv[0], ie = (float)(1.0 / epsv[0]);
        sinkhorn_softmin_pass<<<512, 256, 0, stream>>>(x, y, aL, bL,
            f_ba[1], g_ab[1], f_aa[1], g_bb[1],     // unused in mode 0 (valid mem)
            f_ba[cur], g_ab[cur], f_aa[cur], g_bb[cur],
            e, ie, 0);
    }
    // annealing loop: 4 independent softmins per eps, 0.5-averaged
    for (int it = 0; it < ne; ++it) {
        float e = (float)epsv[it], ie = (float)(1.0 / epsv[it]);
        int nxt = 1 - cur;
        sinkhorn_softmin_pass<<<512, 256, 0, stream>>>(x, y, aL, bL,
            f_ba[cur], g_ab[cur], f_aa[cur], g_bb[cur],
            f_ba[nxt], g_ab[nxt], f_aa[nxt], g_bb[nxt],
            e, ie, 1);
        cur = nxt;
    }
    // final extrapolation at eps = blur^p (overwrite, uses loop potentials)
    {
        float e = (float)epsv[ne - 1], ie = (float)(1.0 / epsv[ne - 1]);
        int nxt = 1 - cur;
        sinkhorn_softmin_pass<<<512, 256, 0, stream>>>(x, y, aL, bL,
            f_ba[cur], g_ab[cur], f_aa[cur], g_bb[cur],
            f_ba[nxt], g_ab[nxt], f_aa[nxt], g_bb[nxt],
            e, ie, 2);
        cur = nxt;
    }

    sinkhorn_reduce_loss<<<1, 256, 0, stream>>>(wx, wy,
        f_ba[cur], f_aa[cur], g_ab[cur], g_bb[cur], out, N);
}